// AttentionAggregation_74526272520923
// MI455X (gfx1250) — compile-verified
//
#include <hip/hip_runtime.h>
#include <math.h>

typedef __attribute__((ext_vector_type(2))) float v2f;
typedef __attribute__((ext_vector_type(8))) float v8f;

#define FDIM 128
#define WPB 8          // waves per block (256 threads, wave32)
#define MAX_BLOCKS 2048

// ---------- helpers ----------

__device__ __forceinline__ void atomic_max_f32(float* addr, float val) {
  // works with init value -inf; int ordering for >=0, uint ordering for <0
  if (val >= 0.0f) atomicMax((int*)addr, __float_as_int(val));
  else             atomicMin((unsigned int*)addr, (unsigned int)__float_as_int(val));
}

// Load the 32 A-fragments (K=128 in steps of 4) for a 16-row tile.
// A 16x4 f32 layout: lanes 0-15 hold M=lane, K=0,1; lanes 16-31 hold M=lane-16, K=2,3.
__device__ __forceinline__ void load_a(const float* __restrict__ X, int row0, int hf, int l,
                                       v2f a[32]) {
  const float* xr = X + (size_t)(row0 + l) * FDIM + 2 * hf;
#pragma unroll
  for (int ks = 0; ks < 32; ++ks) a[ks] = *(const v2f*)(xr + ks * 4);
}

// Two 16x16 output tiles (n-tiles nt0 and nt0+1) in one k-pass.
// WpL index: [nt*1024 + ks*32 + hf*16 + l]. B loads software-pipelined one
// k-step ahead; two independent accumulator chains.
__device__ __forceinline__ void tile_mm_pair(const v2f a[32], const v2f* WpL, int nt0,
                                             int hf, int l, v8f& c0, v8f& c1) {
  const v2f* bp0 = WpL + nt0 * 1024 + hf * 16 + l;
  const v2f* bp1 = bp0 + 1024;
  v2f b0 = bp0[0];
  v2f b1 = bp1[0];
#pragma unroll
  for (int ks = 0; ks < 32; ++ks) {
    v2f nb0, nb1;
    if (ks < 31) {
      nb0 = bp0[(ks + 1) * 32];
      nb1 = bp1[(ks + 1) * 32];
    }
    c0 = __builtin_amdgcn_wmma_f32_16x16x4_f32(false, a[ks], false, b0, (short)0, c0,
                                               false, false);
    c1 = __builtin_amdgcn_wmma_f32_16x16x4_f32(false, a[ks], false, b1, (short)0, c1,
                                               false, false);
    b0 = nb0;
    b1 = nb1;
  }
}

// ---------- kernels ----------

__global__ void init_kernel(float* __restrict__ summed, float* __restrict__ agg,
                            float* __restrict__ nmax, float* __restrict__ nsum, int N) {
  int i = blockIdx.x * blockDim.x + threadIdx.x;
  if (i < N * FDIM) { summed[i] = 0.0f; agg[i] = 0.0f; }
  if (i < N)        { nmax[i] = -INFINITY; nsum[i] = 0.0f; }
}

// Repack row-major W[128][128] into per-lane WMMA B fragments.
// B 4x16 f32 layout: lanes 0-15 -> K=0,1 (v0,v1), N=lane; lanes 16-31 -> K=2,3, N=lane-16.
__global__ void repack_w_kernel(const float* __restrict__ W, v2f* __restrict__ Wp) {
  int i = blockIdx.x * blockDim.x + threadIdx.x;  // 0..8191
  if (i >= 8192) return;
  int l  = i & 15;
  int hf = (i >> 4) & 1;
  int ks = (i >> 5) & 31;
  int nt = i >> 10;
  int k  = ks * 4 + 2 * hf;
  int n  = nt * 16 + l;
  v2f b;
  b.x = W[k * FDIM + n];
  b.y = W[(k + 1) * FDIM + n];
  Wp[i] = b;
}

// summed[recv[e]] += messages[e]  (float4 per thread, 32 chunks per edge row)
__global__ void scatter_sum_kernel(const float* __restrict__ M, const int* __restrict__ recv,
                                   float* __restrict__ summed, int E) {
  int t = blockIdx.x * blockDim.x + threadIdx.x;
  int e = t >> 5;
  if (e >= E) return;
  int c = (t & 31) * 4;
  float4 v = *(const float4*)(M + (size_t)e * FDIM + c);
  float* dst = summed + (size_t)recv[e] * FDIM + c;
  atomicAdd(dst + 0, v.x);
  atomicAdd(dst + 1, v.y);
  atomicAdd(dst + 2, v.z);
  atomicAdd(dst + 3, v.w);
}

// Y = X @ W + bias, X is [ntiles*16, 128]. One wave per 16-row tile, grid-stride.
__global__ void __launch_bounds__(256) gemm_bias_kernel(
    const float* __restrict__ X, const v2f* __restrict__ Wp,
    const float* __restrict__ bias, float* __restrict__ Y, int ntiles) {
  __shared__ v2f WpL[8192];  // 64 KB
  for (int i = threadIdx.x; i < 8192; i += blockDim.x) WpL[i] = Wp[i];
  __syncthreads();

  int lane = threadIdx.x & 31;
  int wave = threadIdx.x >> 5;
  int hf = lane >> 4, l = lane & 15;
  for (int tile = blockIdx.x * WPB + wave; tile < ntiles; tile += gridDim.x * WPB) {
    int row0 = tile * 16;
    v2f a[32];
    load_a(X, row0, hf, l, a);
#pragma unroll
    for (int np = 0; np < 8; np += 2) {
      v8f c0 = {}, c1 = {};
      tile_mm_pair(a, WpL, np, hf, l, c0, c1);
      int n0 = np * 16 + l;
      float bb0 = bias[n0];
      float bb1 = bias[n0 + 16];
#pragma unroll
      for (int r = 0; r < 8; ++r) {
        size_t rowoff = (size_t)(row0 + r + 8 * hf) * FDIM;
        Y[rowoff + n0]      = c0[r] + bb0;
        Y[rowoff + n0 + 16] = c1[r] + bb1;
      }
    }
  }
}

// keys tile (recomputed via WMMA) fused with scores + segment max.
__global__ void __launch_bounds__(256) scores_kernel(
    const float* __restrict__ M, const int* __restrict__ recv,
    const v2f* __restrict__ WkP, const float* __restrict__ bk,
    const float* __restrict__ Q, float* __restrict__ scores,
    float* __restrict__ nmax, int ntiles) {
  __shared__ v2f WpL[8192];
  for (int i = threadIdx.x; i < 8192; i += blockDim.x) WpL[i] = WkP[i];
  __syncthreads();

  int lane = threadIdx.x & 31;
  int wave = threadIdx.x >> 5;
  int hf = lane >> 4, l = lane & 15;
  const float inv_sqrt_d = 0.08838834764831845f;  // 1/sqrt(128)
  for (int tile = blockIdx.x * WPB + wave; tile < ntiles; tile += gridDim.x * WPB) {
    int row0 = tile * 16;
    v2f a[32];
    load_a(M, row0, hf, l, a);
    int rv[8];
#pragma unroll
    for (int r = 0; r < 8; ++r) rv[r] = recv[row0 + r + 8 * hf];
    float sacc[8];
#pragma unroll
    for (int r = 0; r < 8; ++r) sacc[r] = 0.0f;
#pragma unroll
    for (int np = 0; np < 8; np += 2) {
      v8f c0 = {}, c1 = {};
      tile_mm_pair(a, WpL, np, hf, l, c0, c1);
      int n0 = np * 16 + l;
      float bb0 = bk[n0];
      float bb1 = bk[n0 + 16];
#pragma unroll
      for (int r = 0; r < 8; ++r) {
        const float* qr = Q + (size_t)rv[r] * FDIM;
        sacc[r] += (c0[r] + bb0) * qr[n0] + (c1[r] + bb1) * qr[n0 + 16];
      }
    }
#pragma unroll
    for (int r = 0; r < 8; ++r) {
      float s = sacc[r];
      s += __shfl_xor(s, 1, 32);
      s += __shfl_xor(s, 2, 32);
      s += __shfl_xor(s, 4, 32);
      s += __shfl_xor(s, 8, 32);
      s *= inv_sqrt_d;
      if (l == 0) {
        scores[row0 + r + 8 * hf] = s;
        atomic_max_f32(&nmax[rv[r]], s);
      }
    }
  }
}

// exp(score - max) + segment sum of weights
__global__ void softexp_kernel(const float* __restrict__ scores, const int* __restrict__ recv,
                               const float* __restrict__ nmax, float* __restrict__ expw,
                               float* __restrict__ nsum, int E) {
  int e = blockIdx.x * blockDim.x + threadIdx.x;
  if (e >= E) return;
  int r = recv[e];
  float ex = expf(scores[e] - nmax[r]);
  expw[e] = ex;
  atomicAdd(&nsum[r], ex);
}

// values tile (recomputed via WMMA) * attn, scatter-added into agg.
__global__ void __launch_bounds__(256) aggregate_kernel(
    const float* __restrict__ M, const int* __restrict__ recv,
    const v2f* __restrict__ WvP, const float* __restrict__ bv,
    const float* __restrict__ expw, const float* __restrict__ nsum,
    float* __restrict__ agg, int ntiles) {
  __shared__ v2f WpL[8192];
  for (int i = threadIdx.x; i < 8192; i += blockDim.x) WpL[i] = WvP[i];
  __syncthreads();

  int lane = threadIdx.x & 31;
  int wave = threadIdx.x >> 5;
  int hf = lane >> 4, l = lane & 15;
  for (int tile = blockIdx.x * WPB + wave; tile < ntiles; tile += gridDim.x * WPB) {
    int row0 = tile * 16;
    v2f a[32];
    load_a(M, row0, hf, l, a);
    int rv[8];
    float at[8];
#pragma unroll
    for (int r = 0; r < 8; ++r) {
      int m = row0 + r + 8 * hf;
      rv[r] = recv[m];
      at[r] = expw[m] / (nsum[rv[r]] + 1e-8f);
    }
#pragma unroll
    for (int np = 0; np < 8; np += 2) {
      v8f c0 = {}, c1 = {};
      tile_mm_pair(a, WpL, np, hf, l, c0, c1);
      int n0 = np * 16 + l;
      float bb0 = bv[n0];
      float bb1 = bv[n0 + 16];
#pragma unroll
      for (int r = 0; r < 8; ++r) {
        float* ar = agg + (size_t)rv[r] * FDIM;
        atomicAdd(ar + n0,      (c0[r] + bb0) * at[r]);
        atomicAdd(ar + n0 + 16, (c1[r] + bb1) * at[r]);
      }
    }
  }
}

// ---------- host launch ----------

static inline int grid_for(int ntiles) {
  int b = (ntiles + WPB - 1) / WPB;
  return b < MAX_BLOCKS ? b : MAX_BLOCKS;
}

extern "C" void kernel_launch(void* const* d_in, const int* in_sizes, int n_in,
                              void* d_out, int out_size, void* d_ws, size_t ws_size,
                              hipStream_t stream) {
  const float* messages  = (const float*)d_in[0];
  const int*   receivers = (const int*)d_in[1];
  const float* Wk = (const float*)d_in[3];
  const float* bk = (const float*)d_in[4];
  const float* Wv = (const float*)d_in[5];
  const float* bv = (const float*)d_in[6];
  const float* Wq = (const float*)d_in[7];
  const float* bq = (const float*)d_in[8];
  const float* Wo = (const float*)d_in[9];
  const float* bo = (const float*)d_in[10];

  const int E = in_sizes[0] / FDIM;   // 640000
  const int N = out_size / FDIM;      // 40000

  float* ws = (float*)d_ws;
  float* summed  = ws; ws += (size_t)N * FDIM;
  float* queries = ws; ws += (size_t)N * FDIM;
  float* agg     = ws; ws += (size_t)N * FDIM;
  float* nmax    = ws; ws += N;
  float* nsum    = ws; ws += N;
  float* scores  = ws; ws += E;
  float* expw    = ws; ws += E;
  v2f* WkP = (v2f*)ws; ws += 16384;
  v2f* WvP = (v2f*)ws; ws += 16384;
  v2f* WqP = (v2f*)ws; ws += 16384;
  v2f* WoP = (v2f*)ws; ws += 16384;

  const int etiles = E / 16;   // E is a multiple of 16
  const int qtiles = N / 16;   // N is a multiple of 16

  // 1) init node buffers
  init_kernel<<<(N * FDIM + 255) / 256, 256, 0, stream>>>(summed, agg, nmax, nsum, N);

  // 2) repack weight matrices into WMMA B-fragment order
  repack_w_kernel<<<32, 256, 0, stream>>>(Wk, WkP);
  repack_w_kernel<<<32, 256, 0, stream>>>(Wv, WvP);
  repack_w_kernel<<<32, 256, 0, stream>>>(Wq, WqP);
  repack_w_kernel<<<32, 256, 0, stream>>>(Wo, WoP);

  // 3) summed = segment_sum(messages)
  {
    long long tot = (long long)E * 32;
    scatter_sum_kernel<<<(int)((tot + 255) / 256), 256, 0, stream>>>(messages, receivers,
                                                                     summed, E);
  }

  // 4) queries = summed @ Wq + bq
  gemm_bias_kernel<<<grid_for(qtiles), 256, 0, stream>>>(summed, WqP, bq, queries, qtiles);

  // 5) scores + segment max (keys recomputed on the fly)
  scores_kernel<<<grid_for(etiles), 256, 0, stream>>>(messages, receivers, WkP, bk,
                                                      queries, scores, nmax, etiles);

  // 6) exp weights + segment sum
  softexp_kernel<<<(E + 255) / 256, 256, 0, stream>>>(scores, receivers, nmax, expw, nsum, E);

  // 7) agg = segment_sum(values * attn) (values recomputed on the fly)
  aggregate_kernel<<<grid_for(etiles), 256, 0, stream>>>(messages, receivers, WvP, bv,
                                                         expw, nsum, agg, etiles);

  // 8) out = agg @ Wo + bo
  gemm_bias_kernel<<<grid_for(qtiles), 256, 0, stream>>>(agg, WoP, bo, (float*)d_out, qtiles);
}